// S4dLayer_73632919323116
// MI455X (gfx1250) — compile-verified
//
#include <hip/hip_runtime.h>
#include <cstdint>

// S4D layer for MI455X (gfx1250): chunked state-space scan, all matmuls via
// V_WMMA_F32_16X16X4_F32, u staged into LDS with the CDNA5 async global->LDS
// DMA path (ASYNCcnt) overlapped with per-h parameter precompute.
// Memory-bound design (~64MB traffic -> ~2.8us at 23.3 TB/s); WMMA compute
// (~5 GFLOP f32) is negligible.

typedef __attribute__((ext_vector_type(2))) float v2f;
typedef __attribute__((ext_vector_type(8))) float v8f;

#define NB 8
#define NH 512
#define NL 2048
#define NN 64
#define NT 16      // chunk length == WMMA tile
#define NC 128     // chunks per sequence (NL / NT)

__device__ __forceinline__ v8f wmma_f32_16x16x4(v2f a, v2f b, v8f c) {
  // D = A(16x4) * B(4x16) + C(16x16), fp32, wave32
  return __builtin_amdgcn_wmma_f32_16x16x4_f32(
      /*neg_a=*/false, a, /*neg_b=*/false, b,
      /*c_mod=*/(short)0, c, /*reuse_a=*/false, /*reuse_b=*/false);
}

__global__ __launch_bounds__(256)
void s4d_chunked_scan(const float* __restrict__ u,      // (B,H,L)
                      const float* __restrict__ A_re,   // (H,N)
                      const float* __restrict__ A_im,   // (H,N)
                      const float* __restrict__ Cmat,   // (H,N,2)
                      const float* __restrict__ Dvec,   // (H,)
                      const float* __restrict__ logst,  // (H,)
                      float* __restrict__ out)          // (B,H,L)
{
  // Dynamic LDS layout (floats):
  //   uS   [2048]        sequence for this (b,h)
  //   stS  [128*128]     Bst then (in-place) chunk-start states Xcat [c][k]
  //   wtS  [16*128]      WtailCat[s][k] : k<64 -> Re(w_k^(15-s)), else Im
  //   mcS  [128*16]      Mcat[k][t]    : k<64 -> Re(coef_k w^(t+1)), else -Im
  //   tpS  [16*16]       lower-tri Toeplitz of K[0..15]
  //   kS   [16]          K taps (Re sum over n)
  extern __shared__ float smem[];
  float* uS  = smem;              // 2048
  float* stS = uS  + 2048;        // 16384
  float* wtS = stS + 16384;       // 2048
  float* mcS = wtS + 2048;        // 2048
  float* tpS = mcS + 2048;        // 256
  float* kS  = tpS + 256;         // 16

  const int tid = threadIdx.x;
  const int bh  = blockIdx.x;          // b*H + h
  const int h   = bh & (NH - 1);

  // ---- async DMA u -> LDS (GLOBAL_LOAD_ASYNC_TO_LDS_B128, ASYNCcnt) ----
  // 2048 floats = 8192 B; 256 threads x 32 B = two b128 per thread.
  // IOFFSET is added to BOTH the LDS and global addresses, so offset:0/16
  // covers the two 16B halves. Overlaps with the parameter precompute below.
  {
    const float* ub = u + (size_t)bh * NL;
    const uint32_t lds_a = (uint32_t)(uintptr_t)uS + (uint32_t)tid * 32u;
    const uint32_t voff  = (uint32_t)tid * 32u;
    asm volatile(
        "global_load_async_to_lds_b128 %0, %1, %2 offset:0\n\t"
        "global_load_async_to_lds_b128 %0, %1, %2 offset:16"
        :: "v"(lds_a), "v"(voff), "s"(ub) : "memory");
  }

  if (tid < NT) kS[tid] = 0.0f;
  __syncthreads();

  // ---- per-h parameter precompute: lanes 0..63, n = tid ----
  float wTr = 1.0f, wTi = 0.0f;        // w^16 (chunk decay), kept in regs
  if (tid < NN) {
    const int n = tid;
    const float are  = fminf(A_re[h * NN + n], -1e-4f);
    const float aim  = A_im[h * NN + n];
    const float step = __expf(logst[h]);
    const float dre = step * are, dim = step * aim;
    const float er = __expf(dre);
    const float wr = er * __cosf(dim);       // w = exp(dt*A)
    const float wi = er * __sinf(dim);
    const float cre = Cmat[(h * NN + n) * 2 + 0];
    const float cim = Cmat[(h * NN + n) * 2 + 1];
    // coef = C * (w - 1) / A
    const float nr = wr - 1.0f, ni = wi;
    const float den = are * are + aim * aim;
    const float qr = (nr * are + ni * aim) / den;
    const float qi = (ni * are - nr * aim) / den;
    const float cfr = cre * qr - cim * qi;
    const float cfi = cre * qi + cim * qr;
    // powers w^p, p = 0..16
    float pr = 1.0f, pi = 0.0f;
    for (int p = 0; p <= NT; ++p) {
      if (p <= NT - 1) {
        const int s = NT - 1 - p;              // Bst[c,n] = sum_s w^(15-s) u[s]
        wtS[s * 128 + n]      = pr;
        wtS[s * 128 + NN + n] = pi;
        atomicAdd(&kS[p], cfr * pr - cfi * pi);  // K[p] = Re(sum_n coef w^p)
      }
      if (p >= 1) {
        const int t = p - 1;                   // Mcat for Re(coef w^(t+1) X)
        mcS[n * NT + t]        =  (cfr * pr - cfi * pi);
        mcS[(NN + n) * NT + t] = -(cfr * pi + cfi * pr);
      }
      if (p == NT) { wTr = pr; wTi = pi; }
      const float tr = pr * wr - pi * wi;
      const float ti = pr * wi + pi * wr;
      pr = tr; pi = ti;
    }
  }
  __syncthreads();

  // ---- lower-triangular Toeplitz: tp[s][t] = (t>=s) ? K[t-s] : 0 ----
  {
    const int s = tid >> 4, t = tid & 15;
    tpS[tid] = (t >= s) ? kS[t - s] : 0.0f;
  }

  // Drain the async u DMA (per-wave ASYNCcnt), then make it block-visible.
  asm volatile("s_wait_asynccnt 0" ::: "memory");
  __syncthreads();

  // ---- WMMA fragment coordinates (wave32) ----
  const int wave = tid >> 5;          // M-tile: chunks wave*16 .. wave*16+15
  const int lane = tid & 31;
  const int row  = lane & 15;         // A: M index / B,D: N index
  const int half = lane >> 4;         // K/M split between lane halves

  // A fragments: u chunks, A[M=chunk][K=s]. Shared by stages 1 and 4.
  v2f afr[4];
  for (int ks = 0; ks < 4; ++ks) {
    const int s = ks * 4 + 2 * half;
    afr[ks].x = uS[(wave * NT + row) * NT + s];
    afr[ks].y = uS[(wave * NT + row) * NT + s + 1];
  }

  // ---- Stage 1: per-chunk partial states  Bst = U_chunks @ WtailCat ----
  for (int nt = 0; nt < 8; ++nt) {            // 8 N-tiles cover 128 state reals
    v8f acc = {};
    for (int ks = 0; ks < 4; ++ks) {
      const int s = ks * 4 + 2 * half;
      v2f bf;
      bf.x = wtS[s * 128 + nt * NT + row];
      bf.y = wtS[(s + 1) * 128 + nt * NT + row];
      acc = wmma_f32_16x16x4(afr[ks], bf, acc);
    }
    for (int v = 0; v < 8; ++v)               // D[v+8*half][row] -> stS
      stS[(wave * NT + v + 8 * half) * 128 + nt * NT + row] = acc[v];
  }
  __syncthreads();

  // ---- Stage 2: inter-chunk scan  X[c+1] = w^16 * X[c] + Bst[c] ----
  // In place: slot c ends holding the state at the *start* of chunk c.
  if (tid < NN) {
    const int n = tid;
    float xr = 0.0f, xi = 0.0f;
    for (int c = 0; c < NC; ++c) {
      const float br = stS[c * 128 + n];
      const float bi = stS[c * 128 + NN + n];
      stS[c * 128 + n]      = xr;
      stS[c * 128 + NN + n] = xi;
      const float tr = wTr * xr - wTi * xi + br;
      const float ti = wTr * xi + wTi * xr + bi;
      xr = tr; xi = ti;
    }
  }
  __syncthreads();

  // ---- Stage 3: inter-chunk output  Y = Xcat @ Mcat  (K = 128) ----
  v8f y = {};
  for (int ks = 0; ks < 32; ++ks) {
    const int k = ks * 4 + 2 * half;
    v2f af, bf;
    af.x = stS[(wave * NT + row) * 128 + k];
    af.y = stS[(wave * NT + row) * 128 + k + 1];
    bf.x = mcS[k * NT + row];
    bf.y = mcS[(k + 1) * NT + row];
    y = wmma_f32_16x16x4(af, bf, y);
  }

  // ---- Stage 4: intra-chunk causal part  Y += U_chunks @ Toeplitz ----
  for (int ks = 0; ks < 4; ++ks) {
    const int s = ks * 4 + 2 * half;
    v2f bf;
    bf.x = tpS[s * NT + row];
    bf.y = tpS[(s + 1) * NT + row];
    y = wmma_f32_16x16x4(afr[ks], bf, y);
  }

  // ---- Stage 5: epilogue  out = Y + D*u ----
  const float Dh = Dvec[h];
  float* ob = out + (size_t)bh * NL;
  for (int v = 0; v < 8; ++v) {
    const int l = (wave * NT + v + 8 * half) * NT + row;
    ob[l] = y[v] + Dh * uS[l];
  }
}

extern "C" void kernel_launch(void* const* d_in, const int* in_sizes, int n_in,
                              void* d_out, int out_size, void* d_ws, size_t ws_size,
                              hipStream_t stream) {
  const float* u     = (const float*)d_in[0];
  const float* A_re  = (const float*)d_in[1];
  const float* A_im  = (const float*)d_in[2];
  const float* Cmat  = (const float*)d_in[3];
  const float* Dvec  = (const float*)d_in[4];
  const float* logst = (const float*)d_in[5];
  float* out = (float*)d_out;

  const size_t shbytes = (2048 + 16384 + 2048 + 2048 + 256 + 16) * sizeof(float); // ~91 KB
  s4d_chunked_scan<<<dim3(NB * NH), dim3(256), shbytes, stream>>>(
      u, A_re, A_im, Cmat, Dvec, logst, out);
}